// MinGRUBlock_14465449853263
// MI455X (gfx1250) — compile-verified
//
#include <hip/hip_runtime.h>
#include <hip/hip_bf16.h>
#include <math.h>

typedef __attribute__((ext_vector_type(2))) float v2f;
typedef __attribute__((ext_vector_type(4))) float v4f;
typedef __attribute__((ext_vector_type(8))) float v8f;
typedef __attribute__((ext_vector_type(4))) int   v4i;

#define TILE_M 128
#define TILE_N 128
#define TILE_K 16
#define LDSS   20   // LDS row stride in floats: conflict-free & keeps float4 16B-aligned

#if defined(__has_builtin)
#  if __has_builtin(__builtin_amdgcn_global_load_async_to_lds_b128) && \
      __has_builtin(__builtin_amdgcn_s_wait_asynccnt)
#    define USE_ASYNC_LDS 1
#  endif
#endif
#ifndef USE_ASYNC_LDS
#  define USE_ASYNC_LDS 0
#endif

#if USE_ASYNC_LDS
#pragma message("CDNA5: GEMM uses global_load_async_to_lds_b128 + double-buffered LDS")
typedef __attribute__((address_space(1))) v4i gbl_v4i;   // global (AS1) int4
typedef __attribute__((address_space(3))) v4i lds_v4i;   // LDS    (AS3) int4
#else
#pragma message("CDNA5: async-to-LDS builtins not found; GEMM uses sync staging")
#endif

__device__ __forceinline__ float stable_sigmoid(float v) {
    // numerically stable: never exp() a large positive argument
    if (v >= 0.f) { float e = __expf(-v); return 1.f / (1.f + e); }
    float e = __expf(v); return e / (1.f + e);
}

// ---------------------------------------------------------------------------
// Kernel 1: per-row inverse RMS.  grid = B*S blocks, 256 threads.
// ---------------------------------------------------------------------------
__global__ __launch_bounds__(256) void mgru_rms_kernel(
    const float* __restrict__ x, float* __restrict__ invr, int H) {
    const int row = blockIdx.x;
    const float* xr = x + (size_t)row * H;
    float s = 0.f;
    for (int i = threadIdx.x; i < H; i += 256) { float v = xr[i]; s += v * v; }
    __shared__ float red[256];
    red[threadIdx.x] = s;
    __syncthreads();
    for (int off = 128; off > 0; off >>= 1) {
        if (threadIdx.x < off) red[threadIdx.x] += red[threadIdx.x + off];
        __syncthreads();
    }
    if (threadIdx.x == 0) invr[row] = rsqrtf(red[0] / (float)H + 1e-6f);
}

// ---------------------------------------------------------------------------
// Kernel 2: inline RMSNorm * mask -> causal depthwise conv (K=4) + bias ->
// SiLU -> mask.  One thread per (b,s,c) element; coalesced along c.
// ---------------------------------------------------------------------------
__global__ __launch_bounds__(256) void mgru_conv_kernel(
    const float* __restrict__ x, const float* __restrict__ invr,
    const float* __restrict__ mask, const float* __restrict__ nw,
    const float* __restrict__ cw, const float* __restrict__ cb,
    float* __restrict__ h, int S, int H) {
    const int idx = blockIdx.x * 256 + threadIdx.x;
    const int c = idx % H;
    const int s = (idx / H) % S;
    const int b = idx / (H * S);
    const float nwc = nw[c];
    float acc = cb[c];
#pragma unroll
    for (int k = 0; k < 4; ++k) {
        const int sp = s - 3 + k;
        if (sp >= 0) {
            const int r = b * S + sp;
            const float v = x[(size_t)r * H + c] * invr[r] * nwc * mask[r];
            acc += cw[c * 4 + k] * v;
        }
    }
    const float act = acc * stable_sigmoid(acc);   // SiLU
    h[idx] = act * mask[b * S + s];
}

// ---------------------------------------------------------------------------
// Kernel 3: GEMM  C[M,N] = A[M,K] * Bw[N,K]^T  via v_wmma_f32_16x16x4_f32.
// Block tile 128x128, BK=16, 8 waves in 4(M) x 2(N); each wave: 2x4 16x16
// fragments (32x64).  Fragment lane layout per CDNA5 ISA 7.12.2:
//   A 16x4 : lanes 0-15 -> K={0,1}, lanes 16-31 -> K={2,3}
//   B 4x16 : lanes 0-15 -> rows K={0,1}, lanes 16-31 -> rows K={2,3}
//   D 16x16: vgpr r, lane<16 -> M=r, lane>=16 -> M=r+8
// Staging: async global->LDS (ASYNCcnt) with 2-deep LDS double buffer when
// available; otherwise sync b128 loads + ds stores into a single buffer.
// ---------------------------------------------------------------------------
__global__ __launch_bounds__(256) void mgru_gemm_kernel(
    const float* __restrict__ A, const float* __restrict__ Bw,
    float* __restrict__ C, int M, int N, int K) {
#if USE_ASYNC_LDS
    __shared__ float As[2][TILE_M * LDSS];
    __shared__ float Bs[2][TILE_N * LDSS];
#else
    __shared__ float As[1][TILE_M * LDSS];
    __shared__ float Bs[1][TILE_N * LDSS];
#endif

    const int tid   = threadIdx.x;
    const int lane  = tid & 31;
    const int wave  = tid >> 5;      // 0..7
    const int waveM = wave >> 1;     // 0..3  -> 32-row slice
    const int waveN = wave & 1;      // 0..1  -> 64-col slice
    const int r16   = lane & 15;
    const int hi    = lane >> 4;     // 0 or 1

    const int bm = blockIdx.x * TILE_M;
    const int bn = blockIdx.y * TILE_N;

    // per-thread staging coordinates: 512 float4 per 128x16 tile, 2 per thread
    const int f0 = tid, f1 = tid + 256;
    const int row0 = f0 >> 2, col0 = (f0 & 3) << 2;
    const int row1 = f1 >> 2, col1 = (f1 & 3) << 2;
    const float* gA0 = A  + (size_t)(bm + row0) * K + col0;
    const float* gA1 = A  + (size_t)(bm + row1) * K + col1;
    const float* gB0 = Bw + (size_t)(bn + row0) * K + col0;
    const float* gB1 = Bw + (size_t)(bn + row1) * K + col1;

    v8f acc[2][4];
#pragma unroll
    for (int i = 0; i < 2; ++i)
#pragma unroll
        for (int j = 0; j < 4; ++j)
#pragma unroll
            for (int e = 0; e < 8; ++e) acc[i][j][e] = 0.f;

    const int kTiles = K / TILE_K;

#if USE_ASYNC_LDS
    // issue the 4 async 128-bit fills this thread owns for tile kt into buf
    auto issue_tile = [&](int kt, int buf) {
        const int k0 = kt * TILE_K;
        __builtin_amdgcn_global_load_async_to_lds_b128(
            (gbl_v4i*)(gA0 + k0), (lds_v4i*)&As[buf][row0 * LDSS + col0], 0, 0);
        __builtin_amdgcn_global_load_async_to_lds_b128(
            (gbl_v4i*)(gA1 + k0), (lds_v4i*)&As[buf][row1 * LDSS + col1], 0, 0);
        __builtin_amdgcn_global_load_async_to_lds_b128(
            (gbl_v4i*)(gB0 + k0), (lds_v4i*)&Bs[buf][row0 * LDSS + col0], 0, 0);
        __builtin_amdgcn_global_load_async_to_lds_b128(
            (gbl_v4i*)(gB1 + k0), (lds_v4i*)&Bs[buf][row1 * LDSS + col1], 0, 0);
    };
    issue_tile(0, 0);
#endif

    for (int kt = 0; kt < kTiles; ++kt) {
#if USE_ASYNC_LDS
        const int buf = kt & 1;
        if (kt + 1 < kTiles) {
            issue_tile(kt + 1, buf ^ 1);           // overlap fill with compute
            __builtin_amdgcn_s_wait_asynccnt(4);   // our 4 older fills are done
        } else {
            __builtin_amdgcn_s_wait_asynccnt(0);
        }
        __syncthreads();                            // all waves' fills visible
#else
        const int buf = 0;
        const int k0 = kt * TILE_K;
        *(v4f*)&As[0][row0 * LDSS + col0] = *(const v4f*)(gA0 + k0);
        *(v4f*)&As[0][row1 * LDSS + col1] = *(const v4f*)(gA1 + k0);
        *(v4f*)&Bs[0][row0 * LDSS + col0] = *(const v4f*)(gB0 + k0);
        *(v4f*)&Bs[0][row1 * LDSS + col1] = *(const v4f*)(gB1 + k0);
        __syncthreads();
#endif

#pragma unroll
        for (int kk = 0; kk < TILE_K / 4; ++kk) {
            const int kc = kk * 4 + 2 * hi;      // this lane's K pair
            v2f af[2], bf[4];
#pragma unroll
            for (int fm = 0; fm < 2; ++fm)
                af[fm] = *(const v2f*)&As[buf][(waveM * 32 + fm * 16 + r16) * LDSS + kc];
#pragma unroll
            for (int fn = 0; fn < 4; ++fn)
                bf[fn] = *(const v2f*)&Bs[buf][(waveN * 64 + fn * 16 + r16) * LDSS + kc];
#pragma unroll
            for (int fm = 0; fm < 2; ++fm)
#pragma unroll
                for (int fn = 0; fn < 4; ++fn)
                    acc[fm][fn] = __builtin_amdgcn_wmma_f32_16x16x4_f32(
                        false, af[fm], false, bf[fn],
                        (short)0, acc[fm][fn], false, false);
        }
        __syncthreads();   // all waves done reading buf before it is refilled
    }

    // epilogue: D layout -> global
#pragma unroll
    for (int fm = 0; fm < 2; ++fm)
#pragma unroll
        for (int fn = 0; fn < 4; ++fn) {
            const int n  = bn + waveN * 64 + fn * 16 + r16;
            const int mb = bm + waveM * 32 + fm * 16 + hi * 8;
#pragma unroll
            for (int r = 0; r < 8; ++r)
                C[(size_t)(mb + r) * N + n] = acc[fm][fn][r];
        }
}

// ---------------------------------------------------------------------------
// Kernels 4-6: chunked linear MinGRU scan (equivalent to the log-space
// Heinsen scan: h_t = (1-z_t) h_{t-1} + z_t g(hidden_t), h before s=0 is 0).
// ---------------------------------------------------------------------------
__global__ __launch_bounds__(256) void mgru_scan_pass1(
    const float* __restrict__ hg, float* __restrict__ Ac, float* __restrict__ Bv,
    int S, int H, int NC, int CL) {
    const int tid  = blockIdx.x * 256 + threadIdx.x;
    const int c    = tid % H;
    const int rest = tid / H;
    const int ck   = rest % NC;
    const int b    = rest / NC;
    const float* base = hg + (size_t)b * S * 2 * H;
    float Aprod = 1.f, hacc = 0.f;
    const int s0 = ck * CL;
    for (int s = s0; s < s0 + CL; ++s) {
        const float hid = base[(size_t)s * 2 * H + c];
        const float gat = base[(size_t)s * 2 * H + H + c];
        const float z  = stable_sigmoid(gat);
        const float om = stable_sigmoid(-gat);          // 1 - z, stably
        const float g  = (hid >= 0.f) ? hid + 0.5f : stable_sigmoid(hid);
        Aprod *= om;
        hacc = om * hacc + z * g;
    }
    const int o = (b * NC + ck) * H + c;
    Ac[o] = Aprod;
    Bv[o] = hacc;
}

__global__ __launch_bounds__(256) void mgru_scan_pass2(
    const float* __restrict__ Ac, const float* __restrict__ Bv,
    float* __restrict__ Pre, int H, int NC) {
    const int tid = blockIdx.x * 256 + threadIdx.x;
    const int c = tid % H;
    const int b = tid / H;
    float h0 = 0.f;   // padded step t=0 contributes coeff=1, value=0
    for (int ck = 0; ck < NC; ++ck) {
        const int o = (b * NC + ck) * H + c;
        Pre[o] = h0;
        h0 = Ac[o] * h0 + Bv[o];
    }
}

__global__ __launch_bounds__(256) void mgru_scan_pass3(
    const float* __restrict__ hg, const float* __restrict__ Pre,
    const float* __restrict__ x, float* __restrict__ out,
    int S, int H, int NC, int CL) {
    const int tid  = blockIdx.x * 256 + threadIdx.x;
    const int c    = tid % H;
    const int rest = tid / H;
    const int ck   = rest % NC;
    const int b    = rest / NC;
    const float* base = hg + (size_t)b * S * 2 * H;
    float h = Pre[(b * NC + ck) * H + c];
    const int s0 = ck * CL;
    for (int s = s0; s < s0 + CL; ++s) {
        const float hid = base[(size_t)s * 2 * H + c];
        const float gat = base[(size_t)s * 2 * H + H + c];
        const float z  = stable_sigmoid(gat);
        const float om = stable_sigmoid(-gat);
        const float g  = (hid >= 0.f) ? hid + 0.5f : stable_sigmoid(hid);
        h = om * h + z * g;
        const size_t oi = ((size_t)b * S + s) * H + c;
        out[oi] = h + x[oi];
    }
}

// ---------------------------------------------------------------------------
extern "C" void kernel_launch(void* const* d_in, const int* in_sizes, int n_in,
                              void* d_out, int out_size, void* d_ws, size_t ws_size,
                              hipStream_t stream) {
    const float* x    = (const float*)d_in[0];   // [B,S,H]
    const float* mask = (const float*)d_in[1];   // [B,S]
    const float* nw   = (const float*)d_in[2];   // [H]
    const float* cw   = (const float*)d_in[3];   // [H,1,4]
    const float* cb   = (const float*)d_in[4];   // [H]
    const float* whg  = (const float*)d_in[5];   // [2H,H]
    float* out = (float*)d_out;

    constexpr int Bz = 4, S = 2048, H = 2048;
    constexpr int M  = Bz * S;           // 8192 rows
    constexpr int N  = 2 * H;            // 4096 cols
    constexpr int NC = 16, CL = S / NC;  // scan chunks (131072-way parallel)

    float* ws   = (float*)d_ws;
    float* h    = ws;                        // M*H      = 16,777,216 f
    float* hg   = h    + (size_t)M * H;      // M*N      = 33,554,432 f
    float* invr = hg   + (size_t)M * N;      // M        =      8,192 f
    float* Ac   = invr + M;                  // Bz*H*NC  =    131,072 f
    float* Bv   = Ac   + (size_t)Bz * H * NC;
    float* Pre  = Bv   + (size_t)Bz * H * NC;

    mgru_rms_kernel<<<M, 256, 0, stream>>>(x, invr, H);
    mgru_conv_kernel<<<(M * H) / 256, 256, 0, stream>>>(x, invr, mask, nw, cw, cb, h, S, H);
    mgru_gemm_kernel<<<dim3(M / TILE_M, N / TILE_N), 256, 0, stream>>>(h, whg, hg, M, N, H);
    mgru_scan_pass1<<<(Bz * H * NC) / 256, 256, 0, stream>>>(hg, Ac, Bv, S, H, NC, CL);
    mgru_scan_pass2<<<(Bz * H) / 256, 256, 0, stream>>>(Ac, Bv, Pre, H, NC);
    mgru_scan_pass3<<<(Bz * H * NC) / 256, 256, 0, stream>>>(hg, Pre, x, out, S, H, NC, CL);
}